// TreeGRU_30382598652169
// MI455X (gfx1250) — compile-verified
//
#include <hip/hip_runtime.h>
#include <math.h>

typedef __attribute__((ext_vector_type(2))) float v2f;
typedef __attribute__((ext_vector_type(8))) float v8f;

#define DH 128

// ---------------------------------------------------------------------------
// Edge scatter: out[dst[e]] += h[src[e]]          (rn == nullptr)
//               out[dst[e]] += rn[src[e]]*h[src[e]] (rn != nullptr)
// One wave32 per edge; each lane handles 4 consecutive floats (float4 load,
// 4x global_atomic_add_f32). All gather/scatter targets are L2-resident
// (51.2 MB each vs 192 MB L2), so atomics resolve at L2.
// ---------------------------------------------------------------------------
__global__ __launch_bounds__(256) void edge_scatter(
    const float* __restrict__ h, const float* __restrict__ rn,
    const int* __restrict__ src, const int* __restrict__ dst,
    float* __restrict__ outbuf, int E)
{
    int e    = (int)((blockIdx.x * 256u + threadIdx.x) >> 5);
    int lane = threadIdx.x & 31;
    if (e >= E) return;
    int u = src[e];
    int v = dst[e];
    size_t ho = (size_t)u * DH + lane * 4;
    float4 hv = *(const float4*)(h + ho);
    if (rn) {
        float4 rv = *(const float4*)(rn + ho);
        hv.x *= rv.x; hv.y *= rv.y; hv.z *= rv.z; hv.w *= rv.w;
    }
    float* o = outbuf + (size_t)v * DH + lane * 4;
    atomicAdd(o + 0, hv.x);
    atomicAdd(o + 1, hv.y);
    atomicAdd(o + 2, hv.z);
    atomicAdd(o + 3, hv.w);
}

// ---------------------------------------------------------------------------
// Fused dual-GEMM + epilogue:  C = A1@W1 + A2@W2 + bias
//   mode 0: out = sigmoid(C)
//   mode 1: out = (1-z)*s + z*tanh(C)        (z read from zbuf, s from sbuf)
//
// Block = 128 threads = 4 wave32; each wave owns a 16-row stripe and all 128
// output columns (8 WMMA tiles of 16x16 f32, v8f accumulators).
// Weights staged in LDS pre-swizzled to the V_WMMA_F32_16X16X4_F32 B-fragment
// layout: fragment(kstep,tile) -> lane holds {B[k][col], B[k+1][col]} with
// k = 4*kstep + 2*(lane>=16), col = 16*tile + (lane&15).
// Dynamic LDS: 8192 * 8 B = 64 KB (one weight matrix at a time, two phases).
// ---------------------------------------------------------------------------
__device__ __forceinline__ void stage_weights(v2f* ldsB, const float* __restrict__ W,
                                              int tid)
{
    for (int i = tid; i < 8192; i += 128) {
        int p = i >> 7;          // K-pair index 0..63  (K = 2p)
        int n = i & 127;         // output column
        int K = p << 1;
        v2f val;
        val.x = W[K * DH + n];
        val.y = W[(K + 1) * DH + n];
        int kstep = p >> 1;
        int hi    = p & 1;
        int lane  = (hi << 4) | (n & 15);
        int tile  = n >> 4;
        ldsB[(kstep * 8 + tile) * 32 + lane] = val;
    }
}

__device__ __forceinline__ void accum_phase(v8f acc[8], const v2f* ldsB,
                                            const float* __restrict__ A,
                                            int row0, int m, int koff, int lane)
{
    const float* arow = A + (size_t)(row0 + m) * DH;
#pragma unroll 4
    for (int ks = 0; ks < 32; ++ks) {
        v2f a = *(const v2f*)(arow + ks * 4 + koff);
#pragma unroll
        for (int t = 0; t < 8; ++t) {
            v2f bb = ldsB[(ks * 8 + t) * 32 + lane];
            acc[t] = __builtin_amdgcn_wmma_f32_16x16x4_f32(
                false, a, false, bb, (short)0, acc[t], false, false);
        }
    }
}

__global__ __launch_bounds__(128) void gemm_fused(
    const float* __restrict__ A1, const float* __restrict__ W1,
    const float* __restrict__ A2, const float* __restrict__ W2,
    const float* __restrict__ bias, float* __restrict__ out,
    int N, int mode,
    const float* __restrict__ zbuf, const float* __restrict__ sbuf)
{
    extern __shared__ char smem_raw[];
    v2f* ldsB = (v2f*)smem_raw;

    const int tid    = threadIdx.x;
    const int lane   = tid & 31;
    const int waveId = tid >> 5;
    const int m      = lane & 15;          // row within A tile / col within out tile
    const int koff   = (lane >> 4) << 1;   // 0 or 2
    const int row0   = (blockIdx.x * 4 + waveId) * 16;
    const bool active = (row0 < N);        // wave-uniform: EXEC all-1s for WMMA

    v8f acc[8] = {};

    // ---- phase 1: A1 @ W1 ----
    stage_weights(ldsB, W1, tid);
    __syncthreads();
    if (active) accum_phase(acc, ldsB, A1, row0, m, koff, lane);
    __syncthreads();

    // ---- phase 2: A2 @ W2 ----
    stage_weights(ldsB, W2, tid);
    __syncthreads();
    if (active) accum_phase(acc, ldsB, A2, row0, m, koff, lane);

    if (!active) return;

    // ---- epilogue ----
    const int roff = (lane >> 4) << 3;     // rows 0..7 or 8..15 of the tile
#pragma unroll
    for (int t = 0; t < 8; ++t) {
        int col  = t * 16 + m;
        float bv = bias[col];
#pragma unroll
        for (int i = 0; i < 8; ++i) {
            int row = row0 + roff + i;
            size_t idx = (size_t)row * DH + col;
            float c = acc[t][i] + bv;
            if (mode == 0) {
                out[idx] = 1.0f / (1.0f + __expf(-c));
            } else {
                float z  = zbuf[idx];
                float sv = sbuf[idx];
                out[idx] = (1.0f - z) * sv + z * tanhf(c);
            }
        }
    }
}

// ---------------------------------------------------------------------------
// Orchestration:
//   1. s       = scatter_sum(h[src] -> dst)                    [edge pass 1]
//   2. rn      = sigmoid(f_dst@wr + h@ur + br)                 [GEMM, mode 0]
//   3. z(->out)= sigmoid(f_src@wz + s@uz + bz)                 [GEMM, mode 0]
//   4. sum_rh  = scatter_sum(rn[src]*h[src] -> dst)            [edge pass 2]
//   5. out     = (1-z)*s + z*tanh(f_src@w + sum_rh@u + b)      [GEMM, mode 1]
// Workspace: s | rn | sum_rh  (3 * N * 128 floats = 153.6 MB)
// ---------------------------------------------------------------------------
extern "C" void kernel_launch(void* const* d_in, const int* in_sizes, int n_in,
                              void* d_out, int out_size, void* d_ws, size_t ws_size,
                              hipStream_t stream)
{
    const float* h     = (const float*)d_in[0];
    const float* f_src = (const float*)d_in[1];
    const float* f_dst = (const float*)d_in[2];
    const int*   src   = (const int*)d_in[3];
    const int*   dst   = (const int*)d_in[4];
    const float* wz    = (const float*)d_in[5];
    const float* uz    = (const float*)d_in[6];
    const float* bz    = (const float*)d_in[7];
    const float* wr    = (const float*)d_in[8];
    const float* ur    = (const float*)d_in[9];
    const float* br    = (const float*)d_in[10];
    const float* w     = (const float*)d_in[11];
    const float* u     = (const float*)d_in[12];
    const float* b     = (const float*)d_in[13];
    float* out = (float*)d_out;

    const int N = in_sizes[0] / DH;
    const int E = in_sizes[3];
    const size_t NF = (size_t)N * DH;

    float* s_buf   = (float*)d_ws;
    float* rn_buf  = s_buf + NF;
    float* sum_buf = rn_buf + NF;

    hipMemsetAsync(s_buf,   0, NF * sizeof(float), stream);
    hipMemsetAsync(sum_buf, 0, NF * sizeof(float), stream);

    const int edgeBlocks = (E + 7) / 8;               // 8 wave32 per 256-thread block
    const int rowTiles   = (N + 15) / 16;
    const int gemmBlocks = (rowTiles + 3) / 4;        // 4 waves (tiles) per block
    const size_t ldsBytes = 8192 * sizeof(v2f);       // 64 KB

    // 1. s = segment_sum(h[src], dst)
    edge_scatter<<<edgeBlocks, 256, 0, stream>>>(h, nullptr, src, dst, s_buf, E);

    // 2. rn = sigmoid(f_dst@wr + h@ur + br)   (per-node; gathered per-edge later)
    gemm_fused<<<gemmBlocks, 128, ldsBytes, stream>>>(
        f_dst, wr, h, ur, br, rn_buf, N, 0, nullptr, nullptr);

    // 3. z = sigmoid(f_src@wz + s@uz + bz)  -> staged in d_out
    gemm_fused<<<gemmBlocks, 128, ldsBytes, stream>>>(
        f_src, wz, s_buf, uz, bz, out, N, 0, nullptr, nullptr);

    // 4. sum_rh = segment_sum(rn[src]*h[src], dst)
    edge_scatter<<<edgeBlocks, 256, 0, stream>>>(h, rn_buf, src, dst, sum_buf, E);

    // 5. out = (1-z)*s + z*tanh(f_src@w + sum_rh@u + b)
    gemm_fused<<<gemmBlocks, 128, ldsBytes, stream>>>(
        f_src, w, sum_buf, u, b, out, N, 1, out, s_buf);
}